// LSTM_71683004170991
// MI455X (gfx1250) — compile-verified
//
#include <hip/hip_runtime.h>

// Problem constants (from the reference)
#define B_ 4096
#define T_ 512
#define I_ 5
#define H_ 16

typedef __attribute__((ext_vector_type(2))) float v2f;
typedef __attribute__((ext_vector_type(4))) float v4f;
typedef __attribute__((ext_vector_type(8))) float v8f;

// ---------------------------------------------------------------------------
// Fast activations: saturation-safe, short dependent chains (v_exp + v_rcp).
// ---------------------------------------------------------------------------
__device__ __forceinline__ float fast_sigmoid(float x) {
    // 1 / (1 + e^-x); e^-x in [0, inf) -> rcp well-behaved
    return __builtin_amdgcn_rcpf(1.0f + __expf(-x));
}
__device__ __forceinline__ float fast_tanh(float x) {
    // 1 - 2/(e^{2x}+1); e=inf -> 1, e=0 -> -1 (no inf/inf NaN)
    float e = __expf(2.0f * x);
    return 1.0f - 2.0f * __builtin_amdgcn_rcpf(e + 1.0f);
}

// ---------------------------------------------------------------------------
// Kernel 1: input projection for the ONLY batch row that matters (b = B-1).
//   xp[t][j] = sum_i x[B-1,t,i] * W_ih[j,i] + b_ih[j] + b_hh[j]
// 32768 outputs, trivially parallel.
// ---------------------------------------------------------------------------
__global__ __launch_bounds__(256) void lstm_xp_kernel(
    const float* __restrict__ x,
    const float* __restrict__ W_ih,
    const float* __restrict__ b_ih,
    const float* __restrict__ b_hh,
    float* __restrict__ xp) {
    int tid = blockIdx.x * blockDim.x + threadIdx.x;   // 0 .. T*64-1
    int t = tid >> 6;
    int j = tid & 63;
    const float* xr = x + (size_t)(B_ - 1) * T_ * I_ + (size_t)t * I_;
    const float* wr = W_ih + j * I_;
    float acc = b_ih[j] + b_hh[j];
#pragma unroll
    for (int i = 0; i < I_; ++i) acc += xr[i] * wr[i];
    xp[t * 64 + j] = acc;
}

// ---------------------------------------------------------------------------
// Kernel 2: single-wave sequential scan. 32 lanes, wave32.
// Per step: gates(64) = W_hh(64x16) @ h(16) + xp[t]
//   -> four 16x16 gate blocks, each a chain of 4 V_WMMA_F32_16X16X4_F32 with
//      C seeded from xp[t] and B = h-chunk broadcast across all 16 columns.
// A-operand layout (ISA 7.12.2, 32-bit A 16x4):
//   lanes 0-15: M=lane, VGPR0=K0 / VGPR1=K1 ; lanes 16-31: M=lane-16, K2/K3.
// C/D layout: VGPR v = row v (lanes 0-15) / row v+8 (lanes 16-31).
// B layout (4x16): VGPR0 = rows K0 (lanes 0-15) / K2 (lanes 16-31),
//                  VGPR1 = rows K1 / K3  -- h broadcast over N.
// ---------------------------------------------------------------------------
__global__ __launch_bounds__(32) void lstm_scan_kernel(
    const float* __restrict__ xp,
    const float* __restrict__ W_hh,
    const float* __restrict__ W_lin,
    const float* __restrict__ b_lin,
    float* __restrict__ out) {
    const int lane = threadIdx.x;        // 0..31, one wave
    const bool hihalf = lane >= 16;
    const int m = lane & 15;
    const int cbase = hihalf ? 8 : 0;    // C/D row base for this half

    // --- Load W_hh (64x16) once into WMMA A-operand registers: a[block][kchunk]
    v2f a[4][4];
#pragma unroll
    for (int g = 0; g < 4; ++g) {
#pragma unroll
        for (int kc = 0; kc < 4; ++kc) {
            const float* p = W_hh + (16 * g + m) * H_ + 4 * kc + (hihalf ? 2 : 0);
            a[g][kc].x = p[0];
            a[g][kc].y = p[1];
        }
    }

    // State: this half owns h/c for indices cbase..cbase+7 (replicated in 16 lanes)
    float hv[8], cv[8];
#pragma unroll
    for (int v = 0; v < 8; ++v) { hv[v] = 0.0f; cv[v] = 0.0f; }

    // B operands (h chunks broadcast over columns); h0 = 0
    v2f bop[4];
#pragma unroll
    for (int kc = 0; kc < 4; ++kc) { bop[kc].x = 0.0f; bop[kc].y = 0.0f; }

    for (int t = 0; t < T_; ++t) {
        const float* xpt = xp + t * 64;
        v8f d[4];

#pragma unroll
        for (int g = 0; g < 4; ++g) {
            // Seed accumulator with xp[t] gate-block g (bias + input proj ride C)
            v4f clo = *(const v4f*)(xpt + 16 * g + cbase);
            v4f chi = *(const v4f*)(xpt + 16 * g + cbase + 4);
            v8f acc;
            acc[0] = clo.x; acc[1] = clo.y; acc[2] = clo.z; acc[3] = clo.w;
            acc[4] = chi.x; acc[5] = chi.y; acc[6] = chi.z; acc[7] = chi.w;
#pragma unroll
            for (int kc = 0; kc < 4; ++kc) {
                acc = __builtin_amdgcn_wmma_f32_16x16x4_f32(
                    /*neg_a=*/false, a[g][kc],
                    /*neg_b=*/false, bop[kc],
                    /*c_mod=*/(short)0, acc,
                    /*reuse_a=*/false, /*reuse_b=*/false);
            }
            d[g] = acc;
        }

        // Gate order (PyTorch): block 0=i, 1=f, 2=g, 3=o.
        // Lane half owns states s = v + cbase; all columns of D are identical.
#pragma unroll
        for (int v = 0; v < 8; ++v) {
            float ig = fast_sigmoid(d[0][v]);
            float fg = fast_sigmoid(d[1][v]);
            float gg = fast_tanh(d[2][v]);
            float og = fast_sigmoid(d[3][v]);
            float cn = fg * cv[v] + ig * gg;
            cv[v] = cn;
            hv[v] = og * fast_tanh(cn);
        }

        // Broadcast full h(16) to every lane, rebuild B operands for next step
        float hk[16];
#pragma unroll
        for (int k = 0; k < 16; ++k)
            hk[k] = __shfl(hv[k & 7], (k & 8) ? 16 : 0, 32);
#pragma unroll
        for (int kc = 0; kc < 4; ++kc) {
            bop[kc].x = hihalf ? hk[4 * kc + 2] : hk[4 * kc + 0];
            bop[kc].y = hihalf ? hk[4 * kc + 3] : hk[4 * kc + 1];
        }
    }

    // Final head: out[p] = sigmoid(W_lin[p,:] . h + b_lin[p]), p = 0..4
    float hk[16];
#pragma unroll
    for (int k = 0; k < 16; ++k)
        hk[k] = __shfl(hv[k & 7], (k & 8) ? 16 : 0, 32);
    if (lane < 5) {
        float acc = b_lin[lane];
#pragma unroll
        for (int k = 0; k < 16; ++k) acc += W_lin[lane * H_ + k] * hk[k];
        out[lane] = fast_sigmoid(acc);
    }
}

// ---------------------------------------------------------------------------
// Launcher
// ---------------------------------------------------------------------------
extern "C" void kernel_launch(void* const* d_in, const int* in_sizes, int n_in,
                              void* d_out, int out_size, void* d_ws, size_t ws_size,
                              hipStream_t stream) {
    const float* x     = (const float*)d_in[0];
    const float* W_ih  = (const float*)d_in[1];
    const float* W_hh  = (const float*)d_in[2];
    const float* b_ih  = (const float*)d_in[3];
    const float* b_hh  = (const float*)d_in[4];
    const float* W_lin = (const float*)d_in[5];
    const float* b_lin = (const float*)d_in[6];
    float* out = (float*)d_out;
    float* xp  = (float*)d_ws;   // T*64 floats = 128 KB scratch

    lstm_xp_kernel<<<(T_ * 64) / 256, 256, 0, stream>>>(x, W_ih, b_ih, b_hh, xp);
    lstm_scan_kernel<<<1, 32, 0, stream>>>(xp, W_hh, W_lin, b_lin, out);
}